// Sponge_48524540510236
// MI455X (gfx1250) — compile-verified
//
#include <hip/hip_runtime.h>

// 3D wave-equation sponge step, 256^3 fp32.
// Bandwidth-bound (~470 MB/call -> ~20us @ 23.3 TB/s). Strategy:
//  Pass 1: psum = p_cur + source  -> d_out[N..2N) (this IS output #2).
//          psum (64 MiB) then lives in the 192 MB L2 for pass 2's re-reads.
//  Pass 2: z-marching stencil; planes staged into LDS with gfx1250
//          GLOBAL_LOAD_ASYNC_TO_LDS_B32 (ASYNCcnt) double-buffered across
//          4 rotating plane buffers, s_wait_asynccnt + workgroup barrier.

#define GD 256
#define TX 32
#define TY 8
#define PW (TX + 2)            // 34: plane row width incl. x-halo
#define PH (TY + 2)            // 10: plane rows incl. y-halo
#define PLANE (PW * PH)        // 340 floats per staged plane
#define NTHREADS (TX * TY)     // 256 threads = 8 wave32

// ---- gfx1250 async global->LDS copy (4B per lane), guarded ----------------
__device__ __forceinline__ void async_g2l_b32(const float* g, float* l) {
#if __has_builtin(__builtin_amdgcn_global_load_async_to_lds_b32)
  __builtin_amdgcn_global_load_async_to_lds_b32(
      (__attribute__((address_space(1))) int*)(unsigned long long)g,
      (__attribute__((address_space(3))) int*)l,
      /*imm offset*/ 0, /*cpol*/ 0);
#else
  *l = *g;  // synchronous fallback (barrier below still makes this correct)
#endif
}

__device__ __forceinline__ void wait_async_all() {
#if __has_builtin(__builtin_amdgcn_s_wait_asynccnt)
  __builtin_amdgcn_s_wait_asynccnt(0);
#else
  asm volatile("s_wait_asynccnt 0" ::: "memory");
#endif
}

// Stage psum plane `zp` (with x/y halo) into LDS buffer `buf`.
// zp outside [0,GD) -> zero fill (implements the conv zero padding in z).
// interiorXY==true  -> whole 34x10 region in-bounds: pure async path.
// interiorXY==false -> masked synchronous path (x/y zero padding exact).
__device__ __forceinline__ void stage_plane(const float* __restrict__ psum,
                                            float* __restrict__ buf, int zp,
                                            int x0, int y0, bool interiorXY,
                                            int t) {
  if ((unsigned)zp >= (unsigned)GD) {
    for (int e = t; e < PLANE; e += NTHREADS) buf[e] = 0.0f;
    return;
  }
  if (interiorXY) {
    const float* base = psum + ((size_t)zp * GD + (size_t)(y0 - 1)) * GD + (x0 - 1);
    for (int e = t; e < PLANE; e += NTHREADS) {
      int r = e / PW;
      int c = e - r * PW;
      async_g2l_b32(base + (size_t)r * GD + c, &buf[e]);
    }
  } else {
    for (int e = t; e < PLANE; e += NTHREADS) {
      int r = e / PW;
      int c = e - r * PW;
      int gy = y0 - 1 + r;
      int gx = x0 - 1 + c;
      float v = 0.0f;
      if ((unsigned)gx < (unsigned)GD && (unsigned)gy < (unsigned)GD)
        v = psum[((size_t)zp * GD + gy) * GD + gx];
      buf[e] = v;
    }
  }
}

// ---- Pass 1: psum = p_cur + source (also output #2) -----------------------
__global__ void __launch_bounds__(256) sponge_psum_kernel(
    const float4* __restrict__ src, const float4* __restrict__ pc,
    float4* __restrict__ out, int n4) {
  int i = blockIdx.x * blockDim.x + threadIdx.x;
  if (i < n4) {
    float4 a = src[i];
    float4 b = pc[i];
    float4 r;
    r.x = a.x + b.x;
    r.y = a.y + b.y;
    r.z = a.z + b.z;
    r.w = a.w + b.w;
    out[i] = r;
  }
}

// ---- Pass 2: stencil + sponge update --------------------------------------
__global__ void __launch_bounds__(NTHREADS) sponge_stencil_kernel(
    const float* __restrict__ psum, const float* __restrict__ p_pre,
    const float* __restrict__ a, const float* __restrict__ sigma,
    const float* __restrict__ dtp, float* __restrict__ p_new) {
  __shared__ float buf[4][PLANE];  // rotating plane buffers (z-1,z,z+1,z+2)

  const int tx = threadIdx.x, ty = threadIdx.y;
  const int t = ty * TX + tx;
  const int x0 = blockIdx.x * TX;
  const int y0 = blockIdx.y * TY;
  const int x = x0 + tx;
  const int y = y0 + ty;
  const bool interiorXY =
      (x0 >= 1) && (x0 + TX + 1 <= GD) && (y0 >= 1) && (y0 + TY + 1 <= GD);
  const float dt = dtp[0];
  const int ci = (ty + 1) * PW + (tx + 1);  // center index in a plane buffer

  // Prologue: plane -1 (zeros), plane 0, plane 1.
  stage_plane(psum, buf[3], -1, x0, y0, interiorXY, t);
  stage_plane(psum, buf[0], 0, x0, y0, interiorXY, t);
  stage_plane(psum, buf[1], 1, x0, y0, interiorXY, t);
  wait_async_all();
  __syncthreads();

  for (int z = 0; z < GD; ++z) {
    // Prefetch plane z+2 into the buffer not in use this iteration;
    // overlaps with the compute below (waited at loop bottom).
    if (z < GD - 1)
      stage_plane(psum, buf[(z + 2) & 3], z + 2, x0, y0, interiorXY, t);

    const float* bm = buf[(z + 3) & 3];  // plane z-1
    const float* bc = buf[z & 3];        // plane z
    const float* bp = buf[(z + 1) & 3];  // plane z+1

    const float center = bc[ci];
    const float lap = bc[ci - 1] + bc[ci + 1] + bc[ci - PW] + bc[ci + PW] +
                      bm[ci] + bp[ci] - 6.0f * center;

    const size_t idx = ((size_t)z * GD + y) * GD + x;
    const float half = 0.5f * dt * sigma[idx];
    const float pn =
        (2.0f * center - (1.0f - half) * p_pre[idx] + a[idx] * lap) /
        (1.0f + half);
    p_new[idx] = pn;

    wait_async_all();   // plane z+2 LDS writes complete (this wave)
    __syncthreads();    // all waves' portions complete
  }
}

extern "C" void kernel_launch(void* const* d_in, const int* in_sizes, int n_in,
                              void* d_out, int out_size, void* d_ws,
                              size_t ws_size, hipStream_t stream) {
  (void)in_sizes; (void)n_in; (void)out_size; (void)d_ws; (void)ws_size;
  const float* source_cur = (const float*)d_in[0];
  const float* p_cur      = (const float*)d_in[1];
  const float* p_pre      = (const float*)d_in[2];
  const float* a          = (const float*)d_in[3];
  const float* sigma      = (const float*)d_in[4];
  const float* dtp        = (const float*)d_in[5];

  float* out = (float*)d_out;
  const size_t N = (size_t)GD * GD * GD;
  float* psum = out + N;  // output #2 = p_cur + source, also pass-2 input

  const int n4 = (int)(N / 4);
  sponge_psum_kernel<<<n4 / 256, 256, 0, stream>>>(
      (const float4*)source_cur, (const float4*)p_cur, (float4*)psum, n4);

  dim3 grid(GD / TX, GD / TY);   // 8 x 32 = 256 blocks
  dim3 block(TX, TY);            // 256 threads = 8 wave32
  sponge_stencil_kernel<<<grid, block, 0, stream>>>(psum, p_pre, a, sigma,
                                                    dtp, out);
}